// MGE_30099130811046
// MI455X (gfx1250) — compile-verified
//
#include <hip/hip_runtime.h>
#include <math.h>

// ---------------------------------------------------------------------------
// Sobel-magnitude loss for [64,1,512,512] f32 pair -> scalar mean.
//
// Memory floor: 128 MiB read once -> ~5.5us at 23.3 TB/s. The separable Sobel
// vertical pass runs on the CDNA5 matrix pipe as 16x16 banded-matrix products
// (V_WMMA_F32_16X16X4_F32, full f32), with halo rows folded exactly into the
// WMMA C-accumulator seed. VALU is reserved for the sqrt/magnitude/reduction
// chain; the band A-matrices are built once per block into an LDS table so
// waves just ds_load them (the round-1 version burned ~200 VALU/tile
// rebuilding them with v_cmp/v_cndmask chains per image).
// ---------------------------------------------------------------------------

typedef float v2f __attribute__((ext_vector_type(2)));
typedef float v8f __attribute__((ext_vector_type(8)));

#define IMG_H 512
#define IMG_W 512
#define NBATCH 64
#define WAVES_PER_BLOCK 8
// 64 batches * 32x32 tiles of 16x16 = 65536 tiles, 8 per block
#define NBLOCKS ((NBATCH * 32 * 32) / WAVES_PER_BLOCK)   // 8192
#define NTOT (NBATCH * IMG_H * IMG_W)                    // 16777216
#define TROW 20                                          // padded LDS row stride
#define EPS_F 1e-18f

// Horizontal smoothing [1,2,1] at interior column c (0..15) of padded tile row r (0..17)
__device__ __forceinline__ float sx_at(const float* t, int r, int c) {
    return t[r * TROW + c] + 2.0f * t[r * TROW + c + 1] + t[r * TROW + c + 2];
}
// Horizontal difference [-1,0,1]
__device__ __forceinline__ float dx_at(const float* t, int r, int c) {
    return t[r * TROW + c + 2] - t[r * TROW + c];
}
// Vertical band matrices (interior-K restriction; halo handled via C seed)
__device__ __forceinline__ float band_diff(int m, int k) {
    return (k == m + 1 ? 1.0f : 0.0f) - (k == m - 1 ? 1.0f : 0.0f);
}
__device__ __forceinline__ float band_smooth(int m, int k) {
    return (k == m - 1 ? 1.0f : 0.0f) + (k == m ? 2.0f : 0.0f) + (k == m + 1 ? 1.0f : 0.0f);
}

__global__ __launch_bounds__(256) void sobel_loss_partial(
    const float* __restrict__ y_p, const float* __restrict__ y_t,
    float* __restrict__ partial) {
    // Per-wave 18x18 padded raw tiles for both images (stride 20 vs bank conflicts)
    __shared__ float tile[WAVES_PER_BLOCK][2][18 * TROW];
    // Band A-matrix operand table: identical for every wave/tile/image.
    __shared__ v2f aTabD[4][32];   // vertical-difference band, per kc per lane
    __shared__ v2f aTabS[4][32];   // vertical-smoothing band, per kc per lane
    __shared__ float wsum[WAVES_PER_BLOCK];

    const int tid  = threadIdx.x;
    const int wave = tid >> 5;
    const int lane = tid & 31;
    const int col  = lane & 15;   // N index (and A-matrix M index) for this lane
    const int half = lane >> 4;   // 0: lanes 0-15, 1: lanes 16-31

    const int tileId = blockIdx.x * WAVES_PER_BLOCK + wave;
    const int b   = tileId >> 10;          // batch
    const int rem = tileId & 1023;
    const int r0  = (rem >> 5) << 4;       // tile row origin
    const int c0  = (rem & 31) << 4;       // tile col origin

    // ---- build band A-matrix table once per block (1 float2 per thread) ----
    {
        const int tl  = tid & 31;
        const int tkc = (tid >> 5) & 3;
        const int tm  = tl & 15;
        const int tk0 = 4 * tkc + 2 * (tl >> 4);
        v2f a;
        if (tid < 128) {
            a.x = band_diff(tm, tk0);  a.y = band_diff(tm, tk0 + 1);
            aTabD[tkc][tl] = a;
        } else {
            a.x = band_smooth(tm, tk0); a.y = band_smooth(tm, tk0 + 1);
            aTabS[tkc][tl] = a;
        }
    }

    // ---- fill LDS with zero-padded 18x18 halo tiles, row-wise ----
    // Lanes 0..17 each own one column; column mask is loop-invariant and the
    // row-bounds check is wave-uniform (scalar).
    const int  gc   = c0 - 1 + lane;
    const bool cok  = (lane < 18) && ((unsigned)gc < (unsigned)IMG_W);
    for (int img = 0; img < 2; ++img) {
        const float* src = (img == 0 ? y_p : y_t) + (size_t)b * IMG_H * IMG_W;
        __builtin_prefetch(src + (size_t)r0 * IMG_W + c0, 0, 1);  // global_prefetch_b8
        const float* rowp = src + (ptrdiff_t)(r0 - 1) * IMG_W + gc;
        float* dst = &tile[wave][img][lane];
        for (int lr = 0; lr < 18; ++lr) {
            const int gr = r0 - 1 + lr;          // uniform across the wave
            float v = 0.0f;
            if (cok && (unsigned)gr < (unsigned)IMG_H)
                v = rowp[(ptrdiff_t)lr * IMG_W];
            if (lane < 18) dst[lr * TROW] = v;
        }
    }
    __syncthreads();

    // ---- hoist A-matrix operands for this lane (shared by both images) ----
    v2f aD[4], aS[4];
#pragma unroll
    for (int kc = 0; kc < 4; ++kc) {
        aD[kc] = aTabD[kc][lane];
        aS[kc] = aTabS[kc][lane];
    }

    // ---- per-image Sobel magnitude via WMMA vertical pass ----
    float mag[2][8];
    for (int img = 0; img < 2; ++img) {
        const float* t = &tile[wave][img][0];

        // Exact halo-row contributions folded into the accumulator seed:
        // gh[0] += -sx(row -1), gh[15] += +sx(row 16)  (local rows 0 / 17)
        // gv[0] += +dx(row -1), gv[15] += +dx(row 16)
        const int hr = half ? 17 : 0;
        const float sH = sx_at(t, hr, col);
        const float dV = dx_at(t, hr, col);
        v8f ah = {};
        v8f av = {};
        ah[0] = half ? 0.0f : -sH;
        ah[7] = half ? sH   : 0.0f;
        av[0] = half ? 0.0f : dV;
        av[7] = half ? dV   : 0.0f;

#pragma unroll
        for (int kc = 0; kc < 4; ++kc) {
            const int k0 = 4 * kc + 2 * half;   // per-lane K pair {k0, k0+1}
            // B: 4x16 slices of horizontally-filtered tile (interior row = local K+1)
            v2f bS; bS.x = sx_at(t, k0 + 1, col); bS.y = sx_at(t, k0 + 2, col);
            v2f bD; bD.x = dx_at(t, k0 + 1, col); bD.y = dx_at(t, k0 + 2, col);

            ah = __builtin_amdgcn_wmma_f32_16x16x4_f32(
                false, aD[kc], false, bS, (short)0, ah, false, false);
            av = __builtin_amdgcn_wmma_f32_16x16x4_f32(
                false, aS[kc], false, bD, (short)0, av, false, false);
        }

#pragma unroll
        for (int v = 0; v < 8; ++v)
            mag[img][v] = sqrtf(ah[v] * ah[v] + av[v] * av[v] + EPS_F);
    }

    // ---- loss elements + reduction (reference formula, incl. zero-mask) ----
    float lsum = 0.0f;
#pragma unroll
    for (int v = 0; v < 8; ++v) {
        float d  = mag[1][v] - mag[0][v];
        float t2 = d * d;
        lsum += (t2 == 0.0f) ? 0.0f : sqrtf(t2 + EPS_F);
    }
    for (int off = 16; off > 0; off >>= 1)
        lsum += __shfl_xor(lsum, off, 32);
    if (lane == 0) wsum[wave] = lsum;
    __syncthreads();
    if (tid == 0) {
        float s = 0.0f;
        for (int w = 0; w < WAVES_PER_BLOCK; ++w) s += wsum[w];
        partial[blockIdx.x] = s;
    }
}

__global__ __launch_bounds__(256) void reduce_partials(
    const float* __restrict__ partial, float* __restrict__ out) {
    __shared__ float sh[256];
    float s = 0.0f;
    for (int i = threadIdx.x; i < NBLOCKS; i += 256) s += partial[i];
    sh[threadIdx.x] = s;
    __syncthreads();
    for (int stride = 128; stride > 0; stride >>= 1) {
        if (threadIdx.x < stride) sh[threadIdx.x] += sh[threadIdx.x + stride];
        __syncthreads();
    }
    if (threadIdx.x == 0) out[0] = sh[0] * (1.0f / (float)NTOT);
}

extern "C" void kernel_launch(void* const* d_in, const int* in_sizes, int n_in,
                              void* d_out, int out_size, void* d_ws, size_t ws_size,
                              hipStream_t stream) {
    const float* y_p = (const float*)d_in[0];
    const float* y_t = (const float*)d_in[1];
    float* partial = (float*)d_ws;  // NBLOCKS floats = 32 KB of scratch

    sobel_loss_partial<<<NBLOCKS, 256, 0, stream>>>(y_p, y_t, partial);
    reduce_partials<<<1, 256, 0, stream>>>(partial, (float*)d_out);
}